// STHGCN_18983755448574
// MI455X (gfx1250) — compile-verified
//
#include <hip/hip_runtime.h>

typedef __attribute__((ext_vector_type(2))) float v2f;
typedef __attribute__((ext_vector_type(8))) float v8f;

#define N_TRAJ 25000
#define N_CI   50000
#define NTOT   75000
#define D      128
#define P      2000
#define KNBR   16
#define NTILE  125            // P / 16
#define LN_EPS 1e-5f

// ---------------------------------------------------------------------------
// Kernel 1: x[row] = relu( LN(2*v) * ln_w + ln_b + tw*time + sw*dist )
// v = 0 for trajectory rows, checkin_feature for checkin rows.
// One wave (32 lanes) per row, float4 per lane.
// ---------------------------------------------------------------------------
__global__ __launch_bounds__(256) void k_prep(
    const float* __restrict__ ci_feat,
    const float* __restrict__ time_vec, const float* __restrict__ dist_vec,
    const float* __restrict__ ln_w, const float* __restrict__ ln_b,
    const float* __restrict__ tw, const float* __restrict__ sw,
    float* __restrict__ x)
{
    const int row  = blockIdx.x * 8 + (threadIdx.x >> 5);
    const int lane = threadIdx.x & 31;
    if (row >= NTOT) return;

    float4 t = make_float4(0.f, 0.f, 0.f, 0.f);
    if (row >= N_TRAJ) {
        const float4 v = ((const float4*)(ci_feat + (size_t)(row - N_TRAJ) * D))[lane];
        t.x = 2.f * v.x; t.y = 2.f * v.y; t.z = 2.f * v.z; t.w = 2.f * v.w;
    }
    float s1 = t.x + t.y + t.z + t.w;
    float s2 = t.x * t.x + t.y * t.y + t.z * t.z + t.w * t.w;
    #pragma unroll
    for (int off = 16; off; off >>= 1) {
        s1 += __shfl_xor(s1, off);
        s2 += __shfl_xor(s2, off);
    }
    const float mu  = s1 * (1.f / (float)D);
    const float var = fmaxf(s2 * (1.f / (float)D) - mu * mu, 0.f);
    const float rs  = rsqrtf(var + LN_EPS);

    const float4 w4  = ((const float4*)ln_w)[lane];
    const float4 b4  = ((const float4*)ln_b)[lane];
    const float4 tw4 = ((const float4*)tw)[lane];
    const float4 sw4 = ((const float4*)sw)[lane];
    const float4 tv4 = ((const float4*)time_vec)[lane];
    const float4 dv4 = ((const float4*)dist_vec)[lane];

    float4 y;
    y.x = fmaxf((t.x - mu) * rs * w4.x + b4.x + tw4.x * tv4.x + sw4.x * dv4.x, 0.f);
    y.y = fmaxf((t.y - mu) * rs * w4.y + b4.y + tw4.y * tv4.y + sw4.y * dv4.y, 0.f);
    y.z = fmaxf((t.z - mu) * rs * w4.z + b4.z + tw4.z * tv4.z + sw4.z * dv4.z, 0.f);
    y.w = fmaxf((t.w - mu) * rs * w4.w + b4.w + tw4.w * tv4.w + sw4.w * dv4.w, 0.f);
    ((float4*)x)[(size_t)row * 32 + lane] = y;
}

// ---------------------------------------------------------------------------
// Kernel 2: checkin <- trajectory message passing + final channel add.
// h[i] = LN(x_ci[i] + sum_k softmax(sim)_k * x[nbr_k]) * w + b + tw*time + sw*dist
// One wave per checkin row. Lanes 0..15 hold the 16 sims/neighbors.
// ---------------------------------------------------------------------------
__global__ __launch_bounds__(256) void k_msg(
    const float* __restrict__ x,
    const float* __restrict__ ci_sim, const int* __restrict__ ci_nb,
    const float* __restrict__ time_vec, const float* __restrict__ dist_vec,
    const float* __restrict__ ln_w, const float* __restrict__ ln_b,
    const float* __restrict__ tw, const float* __restrict__ sw,
    float* __restrict__ h)
{
    const int row  = blockIdx.x * 8 + (threadIdx.x >> 5);
    const int lane = threadIdx.x & 31;
    if (row >= N_CI) return;

    float s = -3.0e38f;
    int   nb = 0;
    if (lane < KNBR) {
        s  = ci_sim[(size_t)row * KNBR + lane];
        nb = ci_nb[(size_t)row * KNBR + lane];
    }
    float mx = s;
    #pragma unroll
    for (int off = 16; off; off >>= 1) mx = fmaxf(mx, __shfl_xor(mx, off));
    const float e = __expf(s - mx);           // 0 for lanes >= 16
    float se = e;
    #pragma unroll
    for (int off = 16; off; off >>= 1) se += __shfl_xor(se, off);
    const float wgt = e / se;

    float4 acc = make_float4(0.f, 0.f, 0.f, 0.f);
    #pragma unroll
    for (int k = 0; k < KNBR; ++k) {
        const float wk = __shfl(wgt, k);
        const int   nk = __shfl(nb, k);
        const float4 v = ((const float4*)x)[(size_t)nk * 32 + lane];
        acc.x += wk * v.x; acc.y += wk * v.y; acc.z += wk * v.z; acc.w += wk * v.w;
    }
    const float4 q = ((const float4*)x)[(size_t)(N_TRAJ + row) * 32 + lane];
    float4 t;
    t.x = q.x + acc.x; t.y = q.y + acc.y; t.z = q.z + acc.z; t.w = q.w + acc.w;

    float s1 = t.x + t.y + t.z + t.w;
    float s2 = t.x * t.x + t.y * t.y + t.z * t.z + t.w * t.w;
    #pragma unroll
    for (int off = 16; off; off >>= 1) {
        s1 += __shfl_xor(s1, off);
        s2 += __shfl_xor(s2, off);
    }
    const float mu  = s1 * (1.f / (float)D);
    const float var = fmaxf(s2 * (1.f / (float)D) - mu * mu, 0.f);
    const float rs  = rsqrtf(var + LN_EPS);

    const float4 w4  = ((const float4*)ln_w)[lane];
    const float4 b4  = ((const float4*)ln_b)[lane];
    const float4 tw4 = ((const float4*)tw)[lane];
    const float4 sw4 = ((const float4*)sw)[lane];
    const float4 tv4 = ((const float4*)time_vec)[lane];
    const float4 dv4 = ((const float4*)dist_vec)[lane];

    float4 y;
    y.x = (t.x - mu) * rs * w4.x + b4.x + tw4.x * tv4.x + sw4.x * dv4.x;
    y.y = (t.y - mu) * rs * w4.y + b4.y + tw4.y * tv4.y + sw4.y * dv4.y;
    y.z = (t.z - mu) * rs * w4.z + b4.z + tw4.z * tv4.z + sw4.z * dv4.z;
    y.w = (t.w - mu) * rs * w4.w + b4.w + tw4.w * tv4.w + sw4.w * dv4.w;
    ((float4*)h)[(size_t)row * 32 + lane] = y;
}

// ---------------------------------------------------------------------------
// Kernel 3: logits = h[M x 128] @ W[128 x 2000] + bias  via V_WMMA_F32_16X16X4_F32
// 256 threads = 8 waves; block tile 128(M) x 16(N); K in 2 chunks of 64 with
// register double-buffering (chunk-1 global loads issued before chunk-0 WMMAs).
// A-tile LDS stride 68 (68 mod 64 == 4 -> conflict-free fragment reads).
// Fused epilogue: per-row partial logsumexp over the 16-col tile (half-wave
// shfl reduction on the D-fragment layout), logits streamed out nontemporal.
// ---------------------------------------------------------------------------
#define ASTRIDE 68
__global__ __launch_bounds__(256) void k_gemm(
    const float* __restrict__ A,      // h [M x 128]
    const float* __restrict__ W,      // lin_w [128 x 2000]
    const float* __restrict__ Bias,   // lin_b [2000]
    float* __restrict__ out,          // logits [M x 2000]
    float* __restrict__ pm,           // [M x NTILE] partial row max
    float* __restrict__ ps,           // [M x NTILE] partial row sum(exp)
    int M)
{
    __shared__ float As[128 * ASTRIDE];
    __shared__ float Bs[64 * 16];

    const int tid  = threadIdx.x;
    const int lane = tid & 31;
    const int wv   = tid >> 5;
    const int m0   = blockIdx.x * 128;
    const int n0   = blockIdx.y * 16;

    const int col  = lane & 15;            // N within tile / A row (M) within 16
    const int kb   = (lane >> 4) << 1;     // K sub-offset {0,2}
    const int arow = (wv << 4) + col;      // A row within block tile
    const bool tile_valid = (m0 + (wv << 4)) < M;

    // Per-thread staging indices.
    // A: 8 float4 per chunk; idx = tid + i*256 -> row idx>>4, quad idx&15.
    // B: 1 float4 per chunk; thread t -> B row t>>2, quad t&3.
    const int brow = tid >> 2;
    const int bq4  = (tid & 3) * 4;

    v8f acc = {};
    float4 a_reg[8];
    float4 b_reg;

    // ---- stage chunk 0 (kc = 0) -------------------------------------------
    #pragma unroll
    for (int i = 0; i < 8; ++i) {
        const int idx  = tid + i * 256;
        const int r    = idx >> 4;
        const int c4   = idx & 15;
        const int grow = m0 + r;
        a_reg[i] = make_float4(0.f, 0.f, 0.f, 0.f);
        if (grow < M) a_reg[i] = *(const float4*)(A + (size_t)grow * D + c4 * 4);
    }
    b_reg = *(const float4*)(W + (size_t)brow * P + n0 + bq4);
    #pragma unroll
    for (int i = 0; i < 8; ++i) {
        const int idx = tid + i * 256;
        *(float4*)(&As[(idx >> 4) * ASTRIDE + (idx & 15) * 4]) = a_reg[i];
    }
    *(float4*)(&Bs[brow * 16 + bq4]) = b_reg;
    __syncthreads();

    // ---- prefetch chunk 1 (kc = 64) while computing chunk 0 ---------------
    #pragma unroll
    for (int i = 0; i < 8; ++i) {
        const int idx  = tid + i * 256;
        const int r    = idx >> 4;
        const int c4   = idx & 15;
        const int grow = m0 + r;
        a_reg[i] = make_float4(0.f, 0.f, 0.f, 0.f);
        if (grow < M) a_reg[i] = *(const float4*)(A + (size_t)grow * D + 64 + c4 * 4);
    }
    b_reg = *(const float4*)(W + (size_t)(64 + brow) * P + n0 + bq4);

    if (tile_valid) {
        #pragma unroll
        for (int k0 = 0; k0 < 64; k0 += 4) {
            v2f a, b;
            a[0] = As[arow * ASTRIDE + k0 + kb];
            a[1] = As[arow * ASTRIDE + k0 + kb + 1];
            b[0] = Bs[(k0 + kb) * 16 + col];
            b[1] = Bs[(k0 + kb + 1) * 16 + col];
            acc = __builtin_amdgcn_wmma_f32_16x16x4_f32(
                false, a, false, b, (short)0, acc, false, false);
        }
    }
    __syncthreads();

    // ---- commit chunk 1 to LDS and compute --------------------------------
    #pragma unroll
    for (int i = 0; i < 8; ++i) {
        const int idx = tid + i * 256;
        *(float4*)(&As[(idx >> 4) * ASTRIDE + (idx & 15) * 4]) = a_reg[i];
    }
    *(float4*)(&Bs[brow * 16 + bq4]) = b_reg;
    __syncthreads();

    if (tile_valid) {
        #pragma unroll
        for (int k0 = 0; k0 < 64; k0 += 4) {
            v2f a, b;
            a[0] = As[arow * ASTRIDE + k0 + kb];
            a[1] = As[arow * ASTRIDE + k0 + kb + 1];
            b[0] = Bs[(k0 + kb) * 16 + col];
            b[1] = Bs[(k0 + kb + 1) * 16 + col];
            acc = __builtin_amdgcn_wmma_f32_16x16x4_f32(
                false, a, false, b, (short)0, acc, false, false);
        }

        // ---- epilogue: bias, NT store, fused partial logsumexp ------------
        const float bv    = Bias[n0 + col];
        const int   rbase = m0 + (wv << 4) + (lane >> 4) * 8;
        #pragma unroll
        for (int j = 0; j < 8; ++j) {
            const float v = acc[j] + bv;
            // stream logits to HBM without polluting L2 (write-once data)
            __builtin_nontemporal_store(v, out + (size_t)(rbase + j) * P + n0 + col);
            // partial logsumexp across the 16 columns of this row tile
            // (row j+8*(lane>>4) lives in acc[j] across a 16-lane half)
            float mr = v;
            #pragma unroll
            for (int off = 8; off; off >>= 1) mr = fmaxf(mr, __shfl_xor(mr, off));
            float er = __expf(v - mr);
            #pragma unroll
            for (int off = 8; off; off >>= 1) er += __shfl_xor(er, off);
            if (col == 0) {
                pm[(size_t)(rbase + j) * NTILE + blockIdx.y] = mr;
                ps[(size_t)(rbase + j) * NTILE + blockIdx.y] = er;
            }
        }
    }
}

// ---------------------------------------------------------------------------
// Kernel 4/5/6: NLL loss from per-tile partials. One wave per row combines
// 125 (m,s) pairs online, gathers the label logit, double atomic accumulate.
// ---------------------------------------------------------------------------
__global__ void k_zero(double* acc) { *acc = 0.0; }

__global__ __launch_bounds__(256) void k_loss(
    const float* __restrict__ logits,
    const float* __restrict__ pm, const float* __restrict__ ps,
    const int* __restrict__ label, double* __restrict__ acc)
{
    const int row  = blockIdx.x * 8 + (threadIdx.x >> 5);
    const int lane = threadIdx.x & 31;
    if (row >= N_CI) return;

    float m = -3.0e38f, s = 0.f;
    for (int t = lane; t < NTILE; t += 32) {
        const float mt = pm[(size_t)row * NTILE + t];
        const float st = ps[(size_t)row * NTILE + t];
        if (mt > m) { s = s * __expf(m - mt) + st; m = mt; }
        else        { s += st * __expf(mt - m); }
    }
    #pragma unroll
    for (int off = 16; off; off >>= 1) {
        const float m2 = __shfl_xor(m, off);
        const float s2 = __shfl_xor(s, off);
        const float mn = fmaxf(m, m2);
        s = s * __expf(m - mn) + s2 * __expf(m2 - mn);
        m = mn;
    }
    if (lane == 0) {
        const double lse  = (double)m + (double)__logf(s);
        const double term = (double)logits[(size_t)row * P + label[row]] - lse;
        atomicAdd(acc, term);
    }
}

__global__ void k_fin(const double* acc, float* out_loss)
{
    *out_loss = (float)(-(*acc) / (double)N_CI);
}

// ---------------------------------------------------------------------------
extern "C" void kernel_launch(void* const* d_in, const int* in_sizes, int n_in,
                              void* d_out, int out_size, void* d_ws, size_t ws_size,
                              hipStream_t stream)
{
    const float* ci_feat  = (const float*)d_in[0];
    const float* time_vec = (const float*)d_in[1];
    const float* dist_vec = (const float*)d_in[2];
    // d_in[3] edge_attr0: softmax over K=1 == 1.0 -> unused
    const float* ci_sim   = (const float*)d_in[4];
    // d_in[5] tj_sim: dead code (trajectory update never read) -> unused
    const float* ln_w     = (const float*)d_in[6];
    const float* ln_b     = (const float*)d_in[7];
    const float* tw       = (const float*)d_in[8];
    const float* sw       = (const float*)d_in[9];
    const float* lin_w    = (const float*)d_in[10];
    const float* lin_b    = (const float*)d_in[11];
    const int*   ci_nb    = (const int*)d_in[12];
    // d_in[13] tj_neighbors: dead
    const int*   label    = (const int*)d_in[14];
    float* out = (float*)d_out;

    // Workspace layout (phase overlap: xbuf is dead once k_msg finishes, so
    // the logsumexp partial buffers reuse that region):
    //   hbuf   : N_CI*D floats                       (live k_msg -> k_gemm)
    //   shared : max(NTOT*D, 2*N_CI*NTILE) floats    (xbuf, then pm|ps)
    //   acc    : 1 double
    float*  hbuf   = (float*)d_ws;
    float*  shared = hbuf + (size_t)N_CI * D;
    float*  xbuf   = shared;                              // [NTOT x D]
    float*  pm     = shared;                              // [N_CI x NTILE]
    float*  ps     = pm + (size_t)N_CI * NTILE;           // [N_CI x NTILE]
    double* acc    = (double*)(shared + 2 * (size_t)N_CI * NTILE);

    k_prep<<<(NTOT + 7) / 8, 256, 0, stream>>>(ci_feat, time_vec, dist_vec,
                                               ln_w, ln_b, tw, sw, xbuf);
    k_msg<<<(N_CI + 7) / 8, 256, 0, stream>>>(xbuf, ci_sim, ci_nb,
                                              time_vec, dist_vec,
                                              ln_w, ln_b, tw, sw, hbuf);
    dim3 g((N_CI + 127) / 128, NTILE);
    k_gemm<<<g, 256, 0, stream>>>(hbuf, lin_w, lin_b, out, pm, ps, N_CI);
    k_zero<<<1, 1, 0, stream>>>(acc);
    k_loss<<<(N_CI + 7) / 8, 256, 0, stream>>>(out, pm, ps, label, acc);
    k_fin<<<1, 1, 0, stream>>>(acc, out + (size_t)N_CI * P);
}